// AttentionDecoderGRU_74406013436434
// MI455X (gfx1250) — compile-verified
//
#include <hip/hip_runtime.h>
#include <math.h>

#define H 1024
#define V 50257
#define L 64

typedef float v2f __attribute__((ext_vector_type(2)));
typedef float v8f __attribute__((ext_vector_type(8)));

// ---------------------------------------------------------------------------
// Small helper kernels (all trivially bandwidth bound, coalesced)
// ---------------------------------------------------------------------------

// xcat[0:H) = emb[inp]
__global__ void k_embed(const int* __restrict__ inp, const float* __restrict__ emb,
                        float* __restrict__ xcat) {
    int i = blockIdx.x * 256 + threadIdx.x;
    xcat[i] = emb[(size_t)inp[0] * H + i];
}

// logits[l] = dot([emb[inp]; h], W_attn[l,:]) + b_attn[l]   (one block per l)
__global__ void k_attn_logits(const int* __restrict__ inp, const float* __restrict__ emb,
                              const float* __restrict__ h, const float* __restrict__ Wat,
                              const float* __restrict__ bat, float* __restrict__ logits) {
    __shared__ float red[256];
    const int l = blockIdx.x, tid = threadIdx.x;
    const float* er = emb + (size_t)inp[0] * H;
    const float* wr = Wat + (size_t)l * (2 * H);
    float acc = 0.f;
    for (int i = tid; i < 2 * H; i += 256) {
        float xv = (i < H) ? er[i] : h[i - H];
        acc += xv * wr[i];
    }
    red[tid] = acc; __syncthreads();
    for (int s = 128; s > 0; s >>= 1) {
        if (tid < s) red[tid] += red[tid + s];
        __syncthreads();
    }
    if (tid == 0) logits[l] = red[0] + bat[l];
}

// softmax over 64 logits -> attn_w (ws) and d_out tail
__global__ void k_softmax64(const float* __restrict__ logits, float* __restrict__ attnw,
                            float* __restrict__ out_attn) {
    __shared__ float sh[L];
    __shared__ float stats[2];
    int tid = threadIdx.x;                       // blockDim = 64
    sh[tid] = logits[tid];
    __syncthreads();
    if (tid == 0) {
        float m = sh[0];
        for (int i = 1; i < L; ++i) m = fmaxf(m, sh[i]);
        float s = 0.f;
        for (int i = 0; i < L; ++i) s += expf(sh[i] - m);
        stats[0] = m; stats[1] = s;
    }
    __syncthreads();
    float w = expf(sh[tid] - stats[0]) / stats[1];
    attnw[tid] = w;
    out_attn[tid] = w;
}

// xcat[H:2H) = attn_w @ encoder_outputs
__global__ void k_applied(const float* __restrict__ attnw, const float* __restrict__ enc,
                          float* __restrict__ xcat) {
    __shared__ float wsh[L];
    int tid = threadIdx.x;
    if (tid < L) wsh[tid] = attnw[tid];
    __syncthreads();
    int i = blockIdx.x * 256 + tid;
    float acc = 0.f;
    #pragma unroll 8
    for (int l = 0; l < L; ++l) acc += wsh[l] * enc[(size_t)l * H + i];
    xcat[H + i] = acc;
}

// GRU gate pointwise math (PyTorch r,z,n ordering)
__global__ void k_gru(const float* __restrict__ gx, const float* __restrict__ gh,
                      const float* __restrict__ h, float* __restrict__ hn_ws,
                      float* __restrict__ hn_out) {
    int i = blockIdx.x * 256 + threadIdx.x;
    float r = 1.f / (1.f + expf(-(gx[i] + gh[i])));
    float z = 1.f / (1.f + expf(-(gx[H + i] + gh[H + i])));
    float n = tanhf(gx[2 * H + i] + r * gh[2 * H + i]);
    float v = (1.f - z) * n + z * h[i];
    hn_ws[i] = v;
    hn_out[i] = v;
}

// log-sum-exp stats over V logits
__global__ void k_lse(const float* __restrict__ logits, float* __restrict__ stats) {
    __shared__ float red[256];
    int tid = threadIdx.x;
    float m = -INFINITY;
    for (int i = tid; i < V; i += 256) m = fmaxf(m, logits[i]);
    red[tid] = m; __syncthreads();
    for (int s = 128; s > 0; s >>= 1) {
        if (tid < s) red[tid] = fmaxf(red[tid], red[tid + s]);
        __syncthreads();
    }
    float gmax = red[0];
    __syncthreads();
    float sum = 0.f;
    for (int i = tid; i < V; i += 256) sum += expf(logits[i] - gmax);
    red[tid] = sum; __syncthreads();
    for (int s = 128; s > 0; s >>= 1) {
        if (tid < s) red[tid] += red[tid + s];
        __syncthreads();
    }
    if (tid == 0) { stats[0] = gmax; stats[1] = logf(red[0]); }
}

// in-place: logits -> log_softmax
__global__ void k_logp(float* __restrict__ logits, const float* __restrict__ stats) {
    int i = blockIdx.x * 256 + threadIdx.x;
    if (i < V) logits[i] -= (stats[0] + stats[1]);
}

// ---------------------------------------------------------------------------
// WMMA fp32 GEMV:  y[r] = dot(W[r,:], x) + bias[r]   (optional ReLU)
//
// One wave owns 16 output rows (N dim of B). Per 32-wide K chunk:
//   * CDNA5 async global->LDS staging: each lane moves 64 contiguous bytes
//     with 4x global_load_async_to_lds_b128 (ISA adds INST_OFFSET to both
//     the global and LDS address, matching the contiguous 64B per lane).
//     Tracked by ASYNCcnt; s_wait_asynccnt 0 orders the LDS reads below.
//     Tile row stride 36 floats keeps LDS rows 16B aligned and the strided
//     B-fragment reads nearly bank-conflict free.
//   * 8 chained v_wmma_f32_16x16x4_f32; x broadcast into A row M=0, other
//     rows zero, so D row 0 (VGPR0, lanes 0..15) = the 16 partial dots.
// ~256 HBM bytes per WMMA -> matrix pipe fully hidden under the HBM stream.
// ---------------------------------------------------------------------------
#define GW 8      // waves per block
#define TS 36     // LDS tile row stride in floats (144 B, 16B aligned)

__launch_bounds__(256)
__global__ void k_gemv_wmma(const float* __restrict__ W, const float* __restrict__ x,
                            const float* __restrict__ bias, float* __restrict__ y,
                            int R, int K, int do_relu)
{
    __shared__ float xsh[2 * H];
    __shared__ float tilesh[GW][16 * TS];

    const int tid  = threadIdx.x;
    const int wave = tid >> 5;
    const int lane = tid & 31;

    // stage activation vector once per block
    for (int i = tid; i < K; i += 256) xsh[i] = x[i];
    __syncthreads();

    const int r0 = blockIdx.x * (GW * 16) + wave * 16;
    float* tile  = tilesh[wave];

    // coalesced tile staging mapping: 2 lanes per row, 16 contiguous floats each
    int lrow = r0 + (lane >> 1);
    if (lrow >= R) lrow = R - 1;                 // clamp (stores are guarded)
    const float* wsrc = W + (size_t)lrow * K + (lane & 1) * 16;
    float*       tdst = tile + (lane >> 1) * TS + (lane & 1) * 16;
    // LDS byte offset for the async-DMA destination (flat addr low 32 bits
    // are the LDS offset per the gfx1250 aperture mapping)
    const unsigned tdst_lds = (unsigned)(uintptr_t)tdst;

    // B fragment lane mapping (mirrors documented C layout):
    //   lane holds column n = lane&15; VGPR0 -> k = (lane>>4), VGPR1 -> k = 2+(lane>>4)
    const int  n  = lane & 15;
    const int  kh = lane >> 4;     // 0 or 1
    const bool a0 = (n == 0);      // only A row M=0 carries x

    v8f acc = {0.f, 0.f, 0.f, 0.f, 0.f, 0.f, 0.f, 0.f};

    for (int kc = 0; kc < K; kc += 32) {
        const float* gsrc = wsrc + kc;
        // async DMA 64B/lane straight into LDS (no VGPR round-trip)
        asm volatile(
            "global_load_async_to_lds_b128 %0, %1, off\n\t"
            "global_load_async_to_lds_b128 %0, %1, off offset:16\n\t"
            "global_load_async_to_lds_b128 %0, %1, off offset:32\n\t"
            "global_load_async_to_lds_b128 %0, %1, off offset:48\n\t"
            "s_wait_asynccnt 0x0"
            :
            : "v"(tdst_lds), "v"(gsrc)
            : "memory");
        // per-wave tile + wait above: safe to read; no block barrier needed

        #pragma unroll
        for (int s = 0; s < 8; ++s) {
            const int kb = s * 4;
            v2f b, a;
            b.x = tile[n * TS + kb + kh];
            b.y = tile[n * TS + kb + 2 + kh];
            if (a0) {
                a.x = xsh[kc + kb + 2 * kh];
                a.y = xsh[kc + kb + 2 * kh + 1];
            } else {
                a.x = 0.f; a.y = 0.f;
            }
            // D = A(16x4) x B(4x16) + C ; row 0 of D accumulates the GEMV
            acc = __builtin_amdgcn_wmma_f32_16x16x4_f32(
                false, a, false, b, (short)0, acc, false, false);
        }
    }

    // D row M=0 lives in VGPR0 of lanes 0..15 (N = lane)
    if (lane < 16) {
        int r = r0 + lane;
        if (r < R) {
            float val = acc[0] + bias[r];
            if (do_relu) val = fmaxf(val, 0.f);
            y[r] = val;
        }
    }
}

// ---------------------------------------------------------------------------
// launch
// ---------------------------------------------------------------------------
extern "C" void kernel_launch(void* const* d_in, const int* in_sizes, int n_in,
                              void* d_out, int out_size, void* d_ws, size_t ws_size,
                              hipStream_t stream)
{
    const int*   inp  = (const int*)  d_in[0];
    const float* hid  = (const float*)d_in[1];   // [1,1,H] -> H floats
    const float* enc  = (const float*)d_in[2];   // [L,H]
    const float* emb  = (const float*)d_in[3];   // [V,H]
    const float* Wat  = (const float*)d_in[4];   // [L,2H]
    const float* bat  = (const float*)d_in[5];
    const float* Wc   = (const float*)d_in[6];   // [H,2H]
    const float* bc   = (const float*)d_in[7];
    const float* Wih  = (const float*)d_in[8];   // [3H,H]
    const float* Whh  = (const float*)d_in[9];   // [3H,H]
    const float* bih  = (const float*)d_in[10];
    const float* bhh  = (const float*)d_in[11];
    const float* Wout = (const float*)d_in[12];  // [V,H]
    const float* bout = (const float*)d_in[13];

    float* out = (float*)d_out;       // [ logp(V) | h_new(H) | attn_w(L) ]
    float* ws  = (float*)d_ws;

    float* logits64 = ws;             // 64
    float* xcat     = ws + 64;        // 2048  = [embedded ; applied]
    float* xcomb    = xcat + 2 * H;   // 1024  = relu(W_comb @ xcat + b)
    float* gx       = xcomb + H;      // 3072
    float* gh       = gx + 3 * H;     // 3072
    float* hn       = gh + 3 * H;     // 1024
    float* attnw    = hn + H;         // 64
    float* stats    = attnw + 64;     // 2

    float* logp     = out;            // V (logits, then log_softmax in place)
    float* out_h    = out + V;        // H
    float* out_attn = out + V + H;    // L

    k_embed      <<<H / 256, 256, 0, stream>>>(inp, emb, xcat);
    k_attn_logits<<<L, 256, 0, stream>>>(inp, emb, hid, Wat, bat, logits64);
    k_softmax64  <<<1, L, 0, stream>>>(logits64, attnw, out_attn);
    k_applied    <<<H / 256, 256, 0, stream>>>(attnw, enc, xcat);

    k_gemv_wmma<<<(H     + 127) / 128, 256, 0, stream>>>(Wc,   xcat,  bc,   xcomb, H,     2 * H, 1);
    k_gemv_wmma<<<(3 * H + 127) / 128, 256, 0, stream>>>(Wih,  xcomb, bih,  gx,    3 * H, H,     0);
    k_gemv_wmma<<<(3 * H + 127) / 128, 256, 0, stream>>>(Whh,  hid,   bhh,  gh,    3 * H, H,     0);

    k_gru<<<H / 256, 256, 0, stream>>>(gx, gh, hid, hn, out_h);

    k_gemv_wmma<<<(V + 127) / 128, 256, 0, stream>>>(Wout, hn, bout, logp, V, H, 0);

    k_lse <<<1, 256, 0, stream>>>(logp, stats);
    k_logp<<<(V + 255) / 256, 256, 0, stream>>>(logp, stats);
}